// HSignedConv_44624710205654
// MI455X (gfx1250) — compile-verified
//
#include <hip/hip_runtime.h>
#include <hip/hip_bf16.h>

typedef __attribute__((ext_vector_type(2))) float v2f;
typedef __attribute__((ext_vector_type(8))) float v8f;

#define FDIM      64
#define LSTRIDE   66          // padded row stride in LDS (bank-conflict free)
#define MIN_NORM  1e-15f
#define MAXN      (1.0f - 4e-3f)   // (1 - BALL_EPS)/sqrt(c), c=1
#define ART_CLIP  (1.0f - 1e-7f)

// -------- CDNA5 async global->LDS copy (b64 per lane) --------
__device__ __forceinline__ void async_copy_b64(const float* g, float* l) {
#if __has_builtin(__builtin_amdgcn_global_load_async_to_lds_b64)
    typedef int v2i_vs __attribute__((vector_size(2 * sizeof(int))));
    typedef __attribute__((address_space(1))) v2i_vs* gp_t;
    typedef __attribute__((address_space(3))) v2i_vs* lp_t;
    __builtin_amdgcn_global_load_async_to_lds_b64((gp_t)g, (lp_t)l, 0, 0);
#else
    unsigned loff = (unsigned)(uintptr_t)l;   // low 32 bits = LDS byte offset
    asm volatile("global_load_async_to_lds_b64 %0, %1, off"
                 :: "v"(loff), "v"(g) : "memory");
#endif
}

__device__ __forceinline__ void wait_async0() {
#if __has_builtin(__builtin_amdgcn_s_wait_asynccnt)
    __builtin_amdgcn_s_wait_asynccnt(0);
#else
    asm volatile("s_wait_asynccnt 0x0" ::: "memory");
#endif
}

// -------- wave32 full reduction (sum) --------
__device__ __forceinline__ float wred(float v) {
#pragma unroll
    for (int off = 16; off > 0; off >>= 1) v += __shfl_xor(v, off, 32);
    return v;
}

__device__ __forceinline__ float artanh_f(float x) {
    float t = fminf(x, ART_CLIP);          // x >= 0 here
    return 0.5f * logf((1.0f + t) / (1.0f - t));
}

// mobius_matvec scalar factor (incl. proj back onto ball); res = factor * mx
__device__ __forceinline__ float mm_scale(float in2, float mx2) {
    float xn  = fmaxf(sqrtf(in2), MIN_NORM);
    float mxn = fmaxf(sqrtf(mx2), MIN_NORM);
    float s   = tanhf((mxn / xn) * artanh_f(xn)) / mxn;
    if (mx2 == 0.0f) s = 0.0f;             // zero_mx branch of reference
    // proj: ||res|| = |s| * ||mx||
    float n  = fmaxf(fabsf(s) * sqrtf(mx2), MIN_NORM);
    float ps = (n > MAXN) ? (MAXN / n) : 1.0f;
    return s * ps;
}

// ------------------------------------------------------------------
// Phase 1: scatter-sum of x[src] rows into sum[dst], count into cnt.
// One wave per edge; each lane: one b64 load + two f32 atomics.
// ------------------------------------------------------------------
__global__ void hsc_scatter(const float* __restrict__ x,
                            const int* __restrict__ ei,
                            float* __restrict__ sum,
                            float* __restrict__ cnt,
                            int E) {
    int t = blockIdx.x * blockDim.x + threadIdx.x;
    int e = t >> 5;
    if (e >= E) return;
    int lane = t & 31;
    int c = lane * 2;
    int src = ei[e];
    int dst = ei[E + e];
    const float2 v = *(const float2*)&x[(size_t)src * FDIM + c];
    float* d = &sum[(size_t)dst * FDIM + c];
    atomicAdd(d,     v.x);
    atomicAdd(d + 1, v.y);
    if (lane == 0) atomicAdd(&cnt[dst], 1.0f);
}

// ------------------------------------------------------------------
// Phase 2: per 64-node tile:
//   stage1: async global->LDS copy of raw sum rows and x rows
//   stage2: WMMA f32 16x16x4:  mxa = sum @ W^T,  mxb = x @ Wcc^T
//   stage3: hyperbolic elementwise per node (wave per node);
//           the 1/cnt mean-normalization folded in as scalars here.
// ------------------------------------------------------------------
__global__ void __launch_bounds__(256)
hsc_compute(const float* __restrict__ sum,
            const float* __restrict__ cnt,
            const float* __restrict__ x,
            const float* __restrict__ W,     // [64,64] for agg matvec
            const float* __restrict__ Wcc,   // [64,64] for x matvec
            const float* __restrict__ bcc,   // [64]
            float* __restrict__ out,         // [N,128]
            int outoff, int N) {
    __shared__ float sAgg[64 * LSTRIDE];     // raw sums
    __shared__ float sX  [64 * LSTRIDE];
    __shared__ float sMXa[64 * LSTRIDE];
    __shared__ float sMXb[64 * LSTRIDE];

    const int tid   = threadIdx.x;
    const int node0 = blockIdx.x * 64;
    const int ntile = min(64, N - node0);

    // ---- stage 1: async copy into LDS (zeros for out-of-range rows) ----
    for (int p = tid; p < 64 * (FDIM / 2); p += 256) {   // float2 granules
        int r = p >> 5;            // 32 pairs per row
        int c = (p & 31) * 2;
        float* lx = &sX  [r * LSTRIDE + c];
        float* ls = &sAgg[r * LSTRIDE + c];
        if (r < ntile) {
            size_t g = (size_t)(node0 + r) * FDIM + c;
            async_copy_b64(&x[g],   lx);
            async_copy_b64(&sum[g], ls);
        } else {
            lx[0] = 0.0f; lx[1] = 0.0f;
            ls[0] = 0.0f; ls[1] = 0.0f;
        }
    }
    wait_async0();
    __syncthreads();

    // ---- stage 2: WMMA GEMM (EXEC all-ones; branches are wave-uniform) ----
    {
        const int w    = tid >> 5;           // wave id 0..7
        const int lane = tid & 31;
        const int hf   = lane >> 4;          // lane half
        const int lm   = lane & 15;
        const int mat  = w >> 2;             // 0: sum@W^T  1: x@Wcc^T
        const int rt   = w & 3;              // row tile (16 nodes)
        const float* A  = mat ? sX  : sAgg;
        const float* Wg = mat ? Wcc : W;
        float*       O  = mat ? sMXb : sMXa;
        const int arow = rt * 16 + lm;

#pragma unroll
        for (int ct = 0; ct < 4; ++ct) {     // output col tile
            v8f acc = {};
#pragma unroll
            for (int k0 = 0; k0 < FDIM; k0 += 4) {
                const int ka = k0 + 2 * hf;  // f32 16x16x4 A/B lane layout
                v2f a, b;
                a.x = A[arow * LSTRIDE + ka];
                a.y = A[arow * LSTRIDE + ka + 1];
                const int bn = ct * 16 + lm; // B[k][n] = W[n][k]
                b.x = Wg[bn * FDIM + ka];
                b.y = Wg[bn * FDIM + ka + 1];
                acc = __builtin_amdgcn_wmma_f32_16x16x4_f32(
                        false, a, false, b, (short)0, acc, false, false);
            }
            // D layout: vgpr i -> (M=i, lanes 0-15) / (M=8+i, lanes 16-31)
            const int orow = rt * 16 + hf * 8;
            const int ocol = ct * 16 + lm;
#pragma unroll
            for (int i = 0; i < 8; ++i)
                O[(orow + i) * LSTRIDE + ocol] = acc[i];
        }
    }
    __syncthreads();

    // ---- stage 3: hyperbolic elementwise, one wave per node ----
    {
        const int w    = tid >> 5;
        const int lane = tid & 31;
        const int f0 = lane, f1 = lane + 32;

        // hyp_bias = proj(expmap0(bcc)) -- uniform over nodes, once per wave
        float b0 = bcc[f0], b1 = bcc[f1];
        float bn2 = wred(b0 * b0 + b1 * b1);
        float bn  = fmaxf(sqrtf(bn2), MIN_NORM);
        float sb  = tanhf(bn) / bn;
        float eb0 = sb * b0, eb1 = sb * b1;
        float ebn = fmaxf(wred(eb0 * eb0 + eb1 * eb1), 0.0f);
        float ebnorm = fmaxf(sqrtf(ebn), MIN_NORM);
        float pbs = (ebnorm > MAXN) ? (MAXN / ebnorm) : 1.0f;
        float hb0 = pbs * eb0, hb1 = pbs * eb1;
        float y2  = wred(hb0 * hb0 + hb1 * hb1);

        for (int r = w; r < 64; r += 8) {
            if (r >= ntile) continue;        // wave-uniform guard
            float rc = 1.0f / fmaxf(cnt[node0 + r], 1.0f);   // mean factor

            float va0 = sAgg[r * LSTRIDE + f0], va1 = sAgg[r * LSTRIDE + f1];
            float ma0 = rc * sMXa[r * LSTRIDE + f0];         // (sum@W^T)/cnt
            float ma1 = rc * sMXa[r * LSTRIDE + f1];
            float vx0 = sX  [r * LSTRIDE + f0], vx1 = sX  [r * LSTRIDE + f1];
            float mb0 = sMXb[r * LSTRIDE + f0], mb1 = sMXb[r * LSTRIDE + f1];

            float pa  = wred(va0 * va0 + va1 * va1) * rc * rc; // ||agg||^2
            float pma = wred(ma0 * ma0 + ma1 * ma1);           // ||mxa||^2
            float px  = wred(vx0 * vx0 + vx1 * vx1);           // ||x||^2
            float pmb = wred(mb0 * mb0 + mb1 * mb1);           // ||mxb||^2

            // term1 = proj(mobius_matvec(W, agg))
            float s1 = mm_scale(pa, pma);
            float t1_0 = s1 * ma0, t1_1 = s1 * ma1;

            // term2 = proj(mobius_add(proj(mobius_matvec(Wcc, x)), hyp_bias))
            float s2 = mm_scale(px, pmb);
            float r2_0 = s2 * mb0, r2_1 = s2 * mb1;
            float x2 = s2 * s2 * pmb;                  // ||r2||^2 analytically
            float xy = wred(r2_0 * hb0 + r2_1 * hb1);
            float cnum = 1.0f + 2.0f * xy + y2;
            float cden = fmaxf(1.0f + 2.0f * xy + x2 * y2, MIN_NORM);
            float m0 = (cnum * r2_0 + (1.0f - x2) * hb0) / cden;
            float m1 = (cnum * r2_1 + (1.0f - x2) * hb1) / cden;
            float mn = fmaxf(sqrtf(wred(m0 * m0 + m1 * m1)), MIN_NORM);
            float pm = (mn > MAXN) ? (MAXN / mn) : 1.0f;

            size_t o = (size_t)(node0 + r) * 128 + outoff;
            out[o + f0] = t1_0 + pm * m0;
            out[o + f1] = t1_1 + pm * m1;
        }
    }
}

extern "C" void kernel_launch(void* const* d_in, const int* in_sizes, int n_in,
                              void* d_out, int out_size, void* d_ws, size_t ws_size,
                              hipStream_t stream) {
    const float* x        = (const float*)d_in[0];
    const float* W_pos    = (const float*)d_in[1];
    const float* W_pos_cc = (const float*)d_in[2];
    const float* b_pos_cc = (const float*)d_in[3];
    const float* W_neg    = (const float*)d_in[4];
    const float* W_neg_cc = (const float*)d_in[5];
    const float* b_neg_cc = (const float*)d_in[6];
    const int*   pos_ei   = (const int*)d_in[7];
    const int*   neg_ei   = (const int*)d_in[8];
    float* out = (float*)d_out;

    const int N = in_sizes[0] / FDIM;
    const int E = in_sizes[7] / 2;

    float* sumw = (float*)d_ws;                 // [N*64]
    float* cntw = sumw + (size_t)N * FDIM;      // [N]
    const size_t wsbytes = (size_t)N * (FDIM + 1) * sizeof(float);

    dim3 sblk(256), sgrd((unsigned)(((size_t)E * 32 + 255) / 256));
    dim3 cblk(256), cgrd((unsigned)((N + 63) / 64));

    // positive pass -> out[:, 0:64]
    (void)hipMemsetAsync(d_ws, 0, wsbytes, stream);
    hsc_scatter<<<sgrd, sblk, 0, stream>>>(x, pos_ei, sumw, cntw, E);
    hsc_compute<<<cgrd, cblk, 0, stream>>>(sumw, cntw, x, W_pos, W_pos_cc,
                                           b_pos_cc, out, 0, N);

    // negative pass -> out[:, 64:128] (scratch reused; stream-ordered)
    (void)hipMemsetAsync(d_ws, 0, wsbytes, stream);
    hsc_scatter<<<sgrd, sblk, 0, stream>>>(x, neg_ei, sumw, cntw, E);
    hsc_compute<<<cgrd, cblk, 0, stream>>>(sumw, cntw, x, W_neg, W_neg_cc,
                                           b_neg_cc, out, 64, N);
}